// MPNN_6296422056723
// MI455X (gfx1250) — compile-verified
//
#include <hip/hip_runtime.h>
#include <hip/hip_bf16.h>

typedef __attribute__((ext_vector_type(2))) float v2f;
typedef __attribute__((ext_vector_type(8))) float v8f;

#define D_FEAT 128
#define EDGES_PER_BLOCK 8   // 256 threads / 32 lanes

// ---------------------------------------------------------------------------
// Zero a float buffer (float4-vectorized, grid-stride).
// ---------------------------------------------------------------------------
__global__ void __launch_bounds__(256)
mpnn_zero_f4(float4* __restrict__ p, int n4) {
    int i = blockIdx.x * blockDim.x + threadIdx.x;
    int stride = gridDim.x * blockDim.x;
    for (; i < n4; i += stride) {
        p[i] = make_float4(0.f, 0.f, 0.f, 0.f);
    }
}

// ---------------------------------------------------------------------------
// One message-passing round: hout[dst[e]] += hin[src[e]], D=128 features.
// One wave (32 lanes) per edge; each lane moves a float4 (16B) so the source
// row read is a single coalesced global_load_b128, and the scatter is 4
// global_atomic_add_f32 per lane into L2's atomic units (dest buffer is
// 51 MB -> resident in the 192 MB L2, so atomic RMW stays on-chip).
// ---------------------------------------------------------------------------
__global__ void __launch_bounds__(256)
mpnn_scatter_round(const float* __restrict__ hin,
                   const int*   __restrict__ src,
                   const int*   __restrict__ dst,
                   float*       __restrict__ hout,
                   int nE) {
    const int e = blockIdx.x * EDGES_PER_BLOCK + (threadIdx.x >> 5);
    if (e >= nE) return;
    const int lane = threadIdx.x & 31;

    const int s = src[e];
    const int d = dst[e];

    const float4 v = ((const float4*)(hin + (size_t)s * D_FEAT))[lane];
    float* o = hout + (size_t)d * D_FEAT + lane * 4;
    atomicAdd(o + 0, v.x);
    atomicAdd(o + 1, v.y);
    atomicAdd(o + 2, v.z);
    atomicAdd(o + 3, v.w);
}

// ---------------------------------------------------------------------------
// Final dense layer: out = A @ W^T + bias, A:[nN,128] f32, W:[128,128] f32.
// V_WMMA_F32_16X16X4_F32 per 16x16 output tile, K-loop of 32 steps.
//
// Fragment layouts (CDNA5 ISA, wave32):
//   A (16x4, MxK):  lane L holds M = L%16;  VGPR v holds K = 2*(L/16) + v
//   B (4x16, KxN):  lane L holds N = L%16;  VGPR v holds K = 2*(L/16) + v
//   C/D (16x16):    lane L, VGPR r holds M = r + 8*(L/16), N = L%16
// B[k][n] = W^T[k][n] = W[n][k], so each lane's B pair is a contiguous
// float2 from W row (n), k-major -> coalesced.
// 8 waves per block, one 16x16 tile per wave. nN % 16 == 0 (100000/16=6250).
// ---------------------------------------------------------------------------
__global__ void __launch_bounds__(256)
mpnn_gemm_wmma(const float* __restrict__ A,
               const float* __restrict__ W,
               const float* __restrict__ bias,
               float*       __restrict__ out,
               int nTilesM) {
    const int wave = threadIdx.x >> 5;
    const int lane = threadIdx.x & 31;
    const int tile = blockIdx.x * 8 + wave;          // tiles = nTilesM * 8
    const int mt = tile >> 3;                        // M tile (16 rows)
    const int nt = tile & 7;                         // N tile (16 cols of 128)
    if (mt >= nTilesM) return;

    const int rc = lane & 15;                        // M for A-frag, N for B/C
    const int kh = lane >> 4;                        // K-half selector

    const float* arow = A + ((size_t)(mt * 16 + rc)) * D_FEAT + 2 * kh;
    const float* wrow = W + ((size_t)(nt * 16 + rc)) * D_FEAT + 2 * kh;

    v8f acc;
    const float bn = bias[nt * 16 + rc];             // bias broadcast along M
#pragma unroll
    for (int r = 0; r < 8; ++r) acc[r] = bn;

#pragma unroll 8
    for (int k0 = 0; k0 < D_FEAT; k0 += 4) {
        const float2 av = *(const float2*)(arow + k0);
        const float2 bv = *(const float2*)(wrow + k0);
        v2f a, b;
        a[0] = av.x; a[1] = av.y;
        b[0] = bv.x; b[1] = bv.y;
        acc = __builtin_amdgcn_wmma_f32_16x16x4_f32(
            /*neg_a=*/false, a, /*neg_b=*/false, b,
            /*c_mod=*/(short)0, acc, /*reuse_a=*/false, /*reuse_b=*/false);
    }

    float* obase = out + ((size_t)(mt * 16)) * D_FEAT + nt * 16 + rc;
#pragma unroll
    for (int r = 0; r < 8; ++r) {
        obase[(size_t)(r + 8 * kh) * D_FEAT] = acc[r];
    }
}

// ---------------------------------------------------------------------------
// Orchestration. d_in order: h [N*128 f32], src [E i32], dst [E i32],
// W [128*128 f32], b [128 f32]. d_out: [N*128 f32].
// Ping-pong: h_in -> ws -> d_out -> ws -> (GEMM) -> d_out.
// Workspace need: N*128*4 = 51.2 MB.
// ---------------------------------------------------------------------------
extern "C" void kernel_launch(void* const* d_in, const int* in_sizes, int n_in,
                              void* d_out, int out_size, void* d_ws, size_t ws_size,
                              hipStream_t stream) {
    const float* h    = (const float*)d_in[0];
    const int*   src  = (const int*)d_in[1];
    const int*   dst  = (const int*)d_in[2];
    const float* W    = (const float*)d_in[3];
    const float* bias = (const float*)d_in[4];
    float*       out  = (float*)d_out;
    float*       ws   = (float*)d_ws;

    const int nN = in_sizes[0] / D_FEAT;   // 100000
    const int nE = in_sizes[1];            // 1600000

    const int n4       = (nN * D_FEAT) / 4;
    const int zeroGrid = (n4 + 255) / 256;
    const int scatGrid = (nE + EDGES_PER_BLOCK - 1) / EDGES_PER_BLOCK;
    const int nTilesM  = nN / 16;
    const int gemmGrid = nTilesM;          // 8 tiles (waves) per block

    // Round 1: h -> ws
    mpnn_zero_f4<<<zeroGrid, 256, 0, stream>>>((float4*)ws, n4);
    mpnn_scatter_round<<<scatGrid, 256, 0, stream>>>(h, src, dst, ws, nE);

    // Round 2: ws -> out (out reused as scratch; fully overwritten later)
    mpnn_zero_f4<<<zeroGrid, 256, 0, stream>>>((float4*)out, n4);
    mpnn_scatter_round<<<scatGrid, 256, 0, stream>>>(ws, src, dst, out, nE);

    // Round 3: out -> ws
    mpnn_zero_f4<<<zeroGrid, 256, 0, stream>>>((float4*)ws, n4);
    mpnn_scatter_round<<<scatGrid, 256, 0, stream>>>(out, src, dst, ws, nE);

    // Final linear: out = ws @ W^T + bias (WMMA f32 16x16x4)
    mpnn_gemm_wmma<<<gemmGrid, 256, 0, stream>>>(ws, W, bias, out, nTilesM);
}